// Contextual_Encodings_postprocessor_55774445306088
// MI455X (gfx1250) — compile-verified
//
#include <hip/hip_runtime.h>

// CDNA5 / gfx1250, wave32. Segment-mean via one-hot f32 WMMA scatter.
//
// out[s, 0:1024]    = mean over rows i with seg_i==s of x[b_i, 257+t_i, :]
// out[s, 1024:2048] = mean over rows i with seg_i==s of x[b_i, 0, :]   (CLS)
//
// Block per 16-segment tile; deterministic LDS compaction of matching rows;
// segment-sum expressed as one-hot matmul via V_WMMA_F32_16X16X4_F32 (exact
// f32 accumulate). No global atomics -> bitwise deterministic across replays.
//
// List entries carry precomputed element bases (no row decode in hot loop):
//   entry.x = (b*1024 + 257 + t) << 10   (emb element base)
//   entry.y = (b << 20) | m              (CLS element base | local seg id)
// Round-3 change: list padded to a multiple of 4 with sentinel entries
// (m-field = 16, base = slot 0's row) so the gather+WMMA loop carries NO
// tail/bounds logic: per slot it is ds_load_b64 + and/cmp/cndmask (one-hot)
// + select/add (offset) + global_load_b32 + wmma.

typedef __attribute__((ext_vector_type(2))) float v2f;
typedef __attribute__((ext_vector_type(8))) float v8f;

#define N_BATCH   32
#define SEQ       1024
#define SENT_LEN  512
#define CTX_LEN   256
#define LO        (1 + CTX_LEN)          // 257
#define NUM_SEG   2048
#define NROWS     (N_BATCH * SENT_LEN)   // 16384
#define FEAT      2048
#define SEGS_PER_TILE 16
#define CAP       4096                   // list capacity (expected ~128/tile)

static __device__ __forceinline__ int imin(int a, int b) { return a < b ? a : b; }
static __device__ __forceinline__ int imax(int a, int b) { return a > b ? a : b; }

__launch_bounds__(256, 2)
__global__ void seg_mean_wmma_kernel(const float* __restrict__ x,
                                     const int*   __restrict__ seg,
                                     float*       __restrict__ out) {
    __shared__ int2  s_pair[CAP + 4];        // {emb base, cls base | m} (+pad)
    __shared__ int   s_cnt[SEGS_PER_TILE];   // per-segment row counts
    __shared__ float s_den[SEGS_PER_TILE];   // max(count, 1) as float
    __shared__ int   s_toff[256];            // per-thread exclusive offsets
    __shared__ int   s_total;

    const int tid  = threadIdx.x;
    const int base = blockIdx.x * SEGS_PER_TILE;   // first segment of tile

    if (tid < SEGS_PER_TILE) s_cnt[tid] = 0;

    // ---- Phase 1: per-thread match counts (rows strided by 256) ----
    int c = 0;
    for (int i = tid; i < NROWS; i += 256) {
        const int b = i >> 9, t = i & (SENT_LEN - 1);
        const int s = seg[b * SEQ + LO + t];
        if ((unsigned)(s - base) < (unsigned)SEGS_PER_TILE) c++;
    }
    s_toff[tid] = c;
    __syncthreads();
    if (tid == 0) {                        // serial exclusive scan (256 adds)
        int acc = 0;
        for (int k = 0; k < 256; ++k) { int v = s_toff[k]; s_toff[k] = acc; acc += v; }
        s_total = imin(acc, CAP);
    }
    __syncthreads();

    // ---- Phase 2: fill list with precomputed element bases; integer LDS
    // atomics for counts (deterministic final values) ----
    int pos = s_toff[tid];
    for (int i = tid; i < NROWS; i += 256) {
        const int b = i >> 9, t = i & (SENT_LEN - 1);
        const int s = seg[b * SEQ + LO + t];
        const unsigned d = (unsigned)(s - base);
        if (d < (unsigned)SEGS_PER_TILE) {
            if (pos < CAP) {
                int2 e;
                e.x = (b * SEQ + LO + t) << 10;    // emb element base (fits 31b)
                e.y = (b << 20) | (int)d;          // cls element base | m
                s_pair[pos] = e;
            }
            pos++;
            atomicAdd(&s_cnt[d], 1);
        }
    }
    __syncthreads();
    if (tid < SEGS_PER_TILE) s_den[tid] = (float)imax(s_cnt[tid], 1);

    // ---- Pad list to a multiple of 4 with sentinel entries: m-field = 16
    // (never matches n in 0..15 under a 5-bit compare), base = slot 0's row
    // (valid memory; weight is 0 so the value never contributes) ----
    const int total = s_total;                     // after barrier above? no:
    __syncthreads();                               // ensure s_total visible
    const int totalv = s_total;
    if (tid < 3 && totalv > 0) {
        const int idx = totalv + tid;
        if (idx & 3 || tid < ((4 - (totalv & 3)) & 3)) { /* simple: pad 3 always */ }
        int2 p0 = s_pair[0];
        int2 e; e.x = p0.x; e.y = (p0.y & ~15) | 16;
        s_pair[idx] = e;                           // pad 3 slots unconditionally
    }
    __syncthreads();
    (void)total;

    const int ngroups = (totalv + 3) >> 2;         // 4 rows per WMMA (K=4)

    const int lane = tid & 31;
    const int wave = tid >> 5;                     // 8 waves
    const int n    = lane & 15;                    // B/C column & A's M index
    const int hi   = lane >> 4;                    // lane half
    // ISA A-matrix 16x4 f32 layout: VGPR0 holds K=0 (lanes 0-15) / K=2
    // (lanes 16-31); VGPR1 holds K=1 / K=3. M = lane & 15.
    // B-matrix 4x16 assumed mirrored: component .x <-> K=k0, .y <-> K=k0+1.
    const int k0 = hi ? 2 : 0;

    // ---- Main loop: 16 stripes x (8 waves * 16 feats) = 2048 features ----
    for (int stripe = 0; stripe < FEAT / 128; ++stripe) {
        const int fb    = stripe * 128 + wave * 16;  // wave's feature base
        const int isEmb = (fb < 1024);               // wave-uniform per stripe
        const int colOff = isEmb ? (fb + n) : (fb + n - 1024);

        v8f acc = {};
        for (int g = 0; g < ngroups; ++g) {
            const int gi = (g << 2) + k0;
            const int2 p0 = s_pair[gi];              // one ds_load_b64 each
            const int2 p1 = s_pair[gi + 1];

            // One-hot A: 5-bit compare; sentinel entries (m=16) never match
            v2f a;
            a.x = ((p0.y & 31) == n) ? 1.0f : 0.0f;
            a.y = ((p1.y & 31) == n) ? 1.0f : 0.0f;

            // Gather B: select precomputed emb/CLS base, add column offset
            const int b0 = (isEmb ? p0.x : (p0.y & ~31)) + colOff;
            const int b1 = (isEmb ? p1.x : (p1.y & ~31)) + colOff;
            v2f bv;
            bv.x = x[b0];
            bv.y = x[b1];

            // D(16 segs x 16 feats) += A(16x4 one-hot) * B(4 rows x 16 feats)
            acc = __builtin_amdgcn_wmma_f32_16x16x4_f32(
                false, a, false, bv, (short)0, acc, false, false);
        }

        // ---- Writeback: C VGPR j -> M = j + 8*hi, N = n ----
        union { v8f v; float f[8]; } u; u.v = acc;
        #pragma unroll
        for (int j = 0; j < 8; ++j) {
            const int m = j + (hi << 3);
            out[(long)(base + m) * FEAT + fb + n] = u.f[j] / s_den[m];
        }
    }
}

extern "C" void kernel_launch(void* const* d_in, const int* in_sizes, int n_in,
                              void* d_out, int out_size, void* d_ws, size_t ws_size,
                              hipStream_t stream) {
    (void)in_sizes; (void)n_in; (void)out_size; (void)d_ws; (void)ws_size;
    const float* x   = (const float*)d_in[0];
    const int*   seg = (const int*)d_in[1];
    float*       out = (float*)d_out;
    // 2048 segments / 16 per tile = 128 blocks, 256 threads (8 wave32s)
    seg_mean_wmma_kernel<<<128, 256, 0, stream>>>(x, seg, out);
}